// FPSSampler_31945966748026
// MI455X (gfx1250) — compile-verified
//
#include <hip/hip_runtime.h>
#include <stdint.h>

// ---------------------------------------------------------------------------
// Farthest Point Sampling for MI455X (gfx1250).
//
// Design (reasoned from MI455X specs, no runtime available):
//  * FPS is sequential over M=2048 steps; each step needs a full-N min-dist
//    update + argmax. Arithmetic is tiny (~86 GFLOP total) -> the problem is
//    latency/locality bound, not WMMA material (one query point per step
//    would waste 15/16 WMMA columns; VALU FMA is strictly better).
//  * Keep each batch's 1.5MB point cloud resident in LDS by splitting the
//    batch across SEGS=8 workgroups (16384 pts -> 192KB of the 320KB WGP
//    LDS), loaded ONCE via the Tensor Data Mover (tensor_load_to_lds) and
//    s_wait_tensorcnt. All 2048 iterations then run out of LDS + VGPRs.
//  * Cross-WGP argmax each step via L2: monotonic release/acquire atomic
//    counter + double-buffered candidate slots (dist, idx, xyz coords).
//    Candidates carry the winner's coords, so the output gather is free.
//  * min-dist array lives in VGPRs (16 per thread, fully unrolled).
// ---------------------------------------------------------------------------

#define B_       32
#define N_       131072
#define M_       2048
#define SEGS     8
#define SEGPTS   (N_ / SEGS)     // 16384 points per workgroup
#define T_       1024            // threads per workgroup (32 waves)
#define PPT      (SEGPTS / T_)   // 16 points per thread
#define NW       (T_ / 32)       // 32 waves
#define WS_PER_BATCH 1024        // bytes of comm space per batch

typedef uint32_t u32x4 __attribute__((ext_vector_type(4)));
typedef uint32_t u32x8 __attribute__((ext_vector_type(8)));

// One-shot TDM DMA: 2-D tile (tile_w elems x tile_h rows, row stride in
// elements) from global memory into LDS at byte offset lds_byte.
// D# layout per CDNA5 ISA ch.8 (group0: 4 SGPRs, group1: 8 SGPRs).
__device__ __forceinline__ void tdm_load_2d_to_lds(uint32_t lds_byte,
                                                   const void* gptr,
                                                   uint32_t tile_w,
                                                   uint32_t tile_h,
                                                   uint32_t row_stride) {
  uint64_t ga = (uint64_t)(uintptr_t)gptr;
  u32x4 g0;
  g0[0] = 1u;                                               // count=1, user desc
  g0[1] = lds_byte;                                         // lds_addr
  g0[2] = (uint32_t)ga;                                     // global_addr[31:0]
  g0[3] = ((uint32_t)(ga >> 32) & 0x01FFFFFFu) | (2u << 30);// ga[56:32] | type=2
  u32x8 g1;
  g1[0] = (2u << 16);                                       // data_size=4B, no mcast
  g1[1] = (tile_w & 0xFFFFu) << 16;                         // tensor_dim0[15:0]
  g1[2] = ((tile_w >> 16) & 0xFFFFu) | ((tile_h & 0xFFFFu) << 16); // td0 hi | td1 lo
  g1[3] = (tile_w & 0xFFFFu) << 16;                         // td1 hi=0 | tile_dim0
  g1[4] = (tile_h & 0xFFFFu);                               // tile_dim1 | tile_dim2=0
  g1[5] = row_stride;                                       // tensor_dim0_stride lo32
  g1[6] = 0u;                                               // stride hi | td1_stride lo
  g1[7] = 0u;
  asm volatile("tensor_load_to_lds %0, %1" :: "s"(g0), "s"(g1) : "memory");
}

__device__ __forceinline__ float fcomp(const float4& v, int k) {
  return k == 0 ? v.x : (k == 1 ? v.y : (k == 2 ? v.z : v.w));
}

__global__ __launch_bounds__(T_, 1)
void fps_kernel(const float* __restrict__ x, float* __restrict__ out,
                uint8_t* __restrict__ ws) {
  const int seg  = blockIdx.x;       // segment within batch (0..SEGS-1)
  const int b    = blockIdx.y;       // batch
  const int tid  = threadIdx.x;
  const int wave = tid >> 5;
  const int lane = tid & 31;

  // 3 channel rows packed back-to-back, exactly the TDM tile layout.
  __shared__ float    coords[3 * SEGPTS];   // 192KB
  __shared__ float    red_d[NW];
  __shared__ uint32_t red_i[NW];
  __shared__ float    mbox[4];

  const float4* sx4 = (const float4*)(coords);
  const float4* sy4 = (const float4*)(coords + SEGPTS);
  const float4* sz4 = (const float4*)(coords + 2 * SEGPTS);
  const float*  sx  = coords;
  const float*  sy  = coords + SEGPTS;
  const float*  sz  = coords + 2 * SEGPTS;

  const float* xb = x + (size_t)b * 3u * (size_t)N_;
  const uint32_t segbase = (uint32_t)seg * SEGPTS;

  // --- TDM: DMA this segment's x/y/z rows (16384 x 3, row stride N) to LDS.
  if (tid < 32) {  // one wave issues the tensor op (EXEC ignored by TDM)
    uint32_t lds0 = (uint32_t)(uintptr_t)(&coords[0]);
    tdm_load_2d_to_lds(lds0, xb + segbase, (uint32_t)SEGPTS, 3u, (uint32_t)N_);
    __builtin_amdgcn_s_wait_tensorcnt(0);
  }
  __syncthreads();

  // Per-batch comm block in workspace (zeroed by hipMemsetAsync each launch).
  uint32_t* ctr  = (uint32_t*)(ws + (size_t)b * WS_PER_BATCH);
  uint32_t* cand = (uint32_t*)(ws + (size_t)b * WS_PER_BATCH + 128);
  // slot s at parity p: cand + ((p*SEGS + s) * 8) dwords: {d, gi, x, y, z, ...}

  // First selected point is global index 0 (reference seeds at 0).
  float qx = xb[0], qy = xb[N_], qz = xb[2 * N_];
  if (seg == 0 && tid == 0) {
    float* ob = out + (size_t)b * 3u * (size_t)M_;
    ob[0] = qx; ob[M_] = qy; ob[2 * M_] = qz;
  }

  float dist[PPT];
#pragma unroll
  for (int j = 0; j < PPT; ++j) dist[j] = __builtin_inff();

  for (int m = 1; m < M_; ++m) {
    float    best_d = -1.0f;
    uint32_t best_i = 0u;

    // Update min-dist to newest point; track local argmax.
    // Layout: thread handles 4 groups of 4 contiguous points -> b128 LDS loads.
#pragma unroll
    for (int g = 0; g < PPT / 4; ++g) {
      const int v = g * T_ + tid;           // float4 index
      float4 vx = sx4[v];
      float4 vy = sy4[v];
      float4 vz = sz4[v];
#pragma unroll
      for (int k = 0; k < 4; ++k) {
        float dx = fcomp(vx, k) - qx;
        float dy = fcomp(vy, k) - qy;
        float dz = fcomp(vz, k) - qz;
        float d  = __builtin_fmaf(dx, dx, __builtin_fmaf(dy, dy, dz * dz));
        float nd = fminf(dist[g * 4 + k], d);
        dist[g * 4 + k] = nd;
        uint32_t gi = segbase + (uint32_t)(v * 4 + k);
        bool better = (nd > best_d) || (nd == best_d && gi < best_i);
        best_d = better ? nd : best_d;
        best_i = better ? gi : best_i;
      }
    }

    // Wave32 argmax reduction (first-index tie-break, matching jnp.argmax).
#pragma unroll
    for (int off = 16; off > 0; off >>= 1) {
      float    od = __shfl_xor(best_d, off, 32);
      uint32_t oi = (uint32_t)__shfl_xor((int)best_i, off, 32);
      bool better = (od > best_d) || (od == best_d && oi < best_i);
      best_d = better ? od : best_d;
      best_i = better ? oi : best_i;
    }
    if (lane == 0) { red_d[wave] = best_d; red_i[wave] = best_i; }
    __syncthreads();

    if (wave == 0) {
      best_d = red_d[lane];
      best_i = red_i[lane];
#pragma unroll
      for (int off = 16; off > 0; off >>= 1) {
        float    od = __shfl_xor(best_d, off, 32);
        uint32_t oi = (uint32_t)__shfl_xor((int)best_i, off, 32);
        bool better = (od > best_d) || (od == best_d && oi < best_i);
        best_d = better ? od : best_d;
        best_i = better ? oi : best_i;
      }
      if (lane == 0) {
        const int par = m & 1;
        uint32_t* slot = cand + (uint32_t)((par * SEGS + seg) << 3);
        uint32_t li = best_i - segbase;
        __hip_atomic_store(&slot[0], __float_as_uint(best_d), __ATOMIC_RELAXED, __HIP_MEMORY_SCOPE_AGENT);
        __hip_atomic_store(&slot[1], best_i,                  __ATOMIC_RELAXED, __HIP_MEMORY_SCOPE_AGENT);
        __hip_atomic_store(&slot[2], __float_as_uint(sx[li]), __ATOMIC_RELAXED, __HIP_MEMORY_SCOPE_AGENT);
        __hip_atomic_store(&slot[3], __float_as_uint(sy[li]), __ATOMIC_RELAXED, __HIP_MEMORY_SCOPE_AGENT);
        __hip_atomic_store(&slot[4], __float_as_uint(sz[li]), __ATOMIC_RELAXED, __HIP_MEMORY_SCOPE_AGENT);
        // Arrive (release orders the slot stores), then wait for all 8 segs.
        __hip_atomic_fetch_add(ctr, 1u, __ATOMIC_RELEASE, __HIP_MEMORY_SCOPE_AGENT);
        const uint32_t target = (uint32_t)(SEGS * m);
        while (__hip_atomic_load(ctr, __ATOMIC_ACQUIRE, __HIP_MEMORY_SCOPE_AGENT) < target)
          __builtin_amdgcn_s_sleep(2);
        // Global winner across the 8 segments.
        float wd = -1.0f; uint32_t wi = 0u; float wx = 0.f, wy = 0.f, wz = 0.f;
        for (int s = 0; s < SEGS; ++s) {
          uint32_t* sl = cand + (uint32_t)((par * SEGS + s) << 3);
          float    d  = __uint_as_float(__hip_atomic_load(&sl[0], __ATOMIC_RELAXED, __HIP_MEMORY_SCOPE_AGENT));
          uint32_t gi = __hip_atomic_load(&sl[1], __ATOMIC_RELAXED, __HIP_MEMORY_SCOPE_AGENT);
          float    px = __uint_as_float(__hip_atomic_load(&sl[2], __ATOMIC_RELAXED, __HIP_MEMORY_SCOPE_AGENT));
          float    py = __uint_as_float(__hip_atomic_load(&sl[3], __ATOMIC_RELAXED, __HIP_MEMORY_SCOPE_AGENT));
          float    pz = __uint_as_float(__hip_atomic_load(&sl[4], __ATOMIC_RELAXED, __HIP_MEMORY_SCOPE_AGENT));
          bool better = (d > wd) || (d == wd && gi < wi);
          if (better) { wd = d; wi = gi; wx = px; wy = py; wz = pz; }
        }
        mbox[0] = wx; mbox[1] = wy; mbox[2] = wz;
        if (seg == 0) {   // winner coords ARE the gathered output for step m
          float* ob = out + (size_t)b * 3u * (size_t)M_;
          ob[m] = wx; ob[M_ + m] = wy; ob[2 * M_ + m] = wz;
        }
      }
    }
    __syncthreads();
    qx = mbox[0]; qy = mbox[1]; qz = mbox[2];
  }
}

extern "C" void kernel_launch(void* const* d_in, const int* in_sizes, int n_in,
                              void* d_out, int out_size, void* d_ws, size_t ws_size,
                              hipStream_t stream) {
  (void)in_sizes; (void)n_in; (void)out_size; (void)ws_size;
  const float* x = (const float*)d_in[0];
  float* out = (float*)d_out;
  // Zero the per-batch counters/candidate slots -> deterministic per replay.
  hipMemsetAsync(d_ws, 0, (size_t)B_ * WS_PER_BATCH, stream);
  dim3 grid(SEGS, B_);
  fps_kernel<<<grid, dim3(T_), 0, stream>>>(x, out, (uint8_t*)d_ws);
}